// UnifiedMemory_11287174054578
// MI455X (gfx1250) — compile-verified
//
#include <hip/hip_runtime.h>
#include <math.h>

#define N_BANK   131072
#define D_FEAT   1024
#define BATCH    64
#define MOMENTUM 0.2f
#define INV_TEMP 20.0f   // 1 / 0.05

typedef __attribute__((ext_vector_type(2))) float v2f;
typedef __attribute__((ext_vector_type(8))) float v8f;

// ---------------------------------------------------------------------------
// Kernel 1: L2-normalize each input row -> workspace  (64 rows x 1024)
// ---------------------------------------------------------------------------
__global__ void __launch_bounds__(256)
normalize_rows(const float* __restrict__ in, float* __restrict__ out) {
  __shared__ float red[256];
  const int row = blockIdx.x;
  const float* src = in + (size_t)row * D_FEAT;
  float s = 0.0f;
  for (int i = threadIdx.x; i < D_FEAT; i += 256) {
    float v = src[i];
    s += v * v;
  }
  red[threadIdx.x] = s;
  __syncthreads();
  for (int st = 128; st > 0; st >>= 1) {
    if (threadIdx.x < st) red[threadIdx.x] += red[threadIdx.x + st];
    __syncthreads();
  }
  const float inv = 1.0f / fmaxf(sqrtf(red[0]), 1e-12f);
  float* dst = out + (size_t)row * D_FEAT;
  for (int i = threadIdx.x; i < D_FEAT; i += 256) dst[i] = src[i] * inv;
}

// ---------------------------------------------------------------------------
// Kernel 2: sim = (Xn @ F^T) * 20  via V_WMMA_F32_16X16X4_F32
//   block = 256 threads = 8 waves; wave w: M-tile = (w&3)*16, N-tile = blk*32+(w>>2)*16
//   A frag (16x4 f32): lane<16 -> row lane, k..k+1 ; lane>=16 -> row lane-16, k+2..k+3
//   B frag (4x16 f32): same pattern over F rows (B[k][n] = F[n][k])
//   D: VGPR j -> m = mbase + j + 8*(lane>=16), n = nbase + (lane&15)
// ---------------------------------------------------------------------------
__global__ void __launch_bounds__(256)
sim_gemm(const float* __restrict__ Xn, const float* __restrict__ F,
         float* __restrict__ sim) {
  const int lane  = threadIdx.x & 31;
  const int w     = threadIdx.x >> 5;        // 0..7
  const int mbase = (w & 3) * 16;
  const int nbase = blockIdx.x * 32 + (w >> 2) * 16;
  const int half  = lane >> 4;               // 0 or 1
  const int r     = lane & 15;

  const float* arow = Xn + (size_t)(mbase + r) * D_FEAT + half * 2;
  const float* brow = F  + (size_t)(nbase + r) * D_FEAT + half * 2;

  v8f acc = {};
  for (int kk = 0; kk < D_FEAT; kk += 128) {
    if (kk + 128 < D_FEAT) {
      __builtin_prefetch(brow + kk + 128, 0, 1);  // global_prefetch on streamed bank
    }
#pragma unroll
    for (int k = kk; k < kk + 128; k += 4) {
      v2f a = *(const v2f*)(arow + k);
      v2f b = *(const v2f*)(brow + k);
      // (neg_a, A, neg_b, B, c_mod, C, reuse_a, reuse_b)
      acc = __builtin_amdgcn_wmma_f32_16x16x4_f32(false, a, false, b,
                                                  (short)0, acc, false, false);
    }
  }

  float* out = sim + (size_t)(nbase + r);
#pragma unroll
  for (int j = 0; j < 8; ++j) {
    const int m = mbase + j + half * 8;
    out[(size_t)m * N_BANK] = acc[j] * INV_TEMP;
  }
}

// ---------------------------------------------------------------------------
// Kernel 3: copy bank -> output (float4 vectorized, grid-stride)
// ---------------------------------------------------------------------------
__global__ void copy_bank(const float4* __restrict__ src, float4* __restrict__ dst,
                          size_t n4) {
  size_t i      = (size_t)blockIdx.x * blockDim.x + threadIdx.x;
  size_t stride = (size_t)gridDim.x * blockDim.x;
  for (; i < n4; i += stride) dst[i] = src[i];
}

// ---------------------------------------------------------------------------
// Kernel 4: sequential momentum scatter-update (+renorm), exact scan order.
//   Single block of 1024 threads, one thread per feature element.
// ---------------------------------------------------------------------------
__global__ void __launch_bounds__(1024)
momentum_update(const float* __restrict__ Xn, const int* __restrict__ idx,
                float* __restrict__ feat_out) {
  __shared__ float red[1024];
  const int t = threadIdx.x;
  for (int i = 0; i < BATCH; ++i) {
    const int y = idx[i];
    float* row = feat_out + (size_t)y * D_FEAT;
    const float v = MOMENTUM * row[t] + (1.0f - MOMENTUM) * Xn[(size_t)i * D_FEAT + t];
    red[t] = v * v;
    __syncthreads();
    for (int s = 512; s > 0; s >>= 1) {
      if (t < s) red[t] += red[t + s];
      __syncthreads();
    }
    const float inv = 1.0f / fmaxf(sqrtf(red[0]), 1e-12f);
    __syncthreads();           // protect red[] before next iteration overwrites
    row[t] = v * inv;
    __syncthreads();           // writes visible before next iteration's reads
  }
}

// ---------------------------------------------------------------------------
extern "C" void kernel_launch(void* const* d_in, const int* in_sizes, int n_in,
                              void* d_out, int out_size, void* d_ws, size_t ws_size,
                              hipStream_t stream) {
  const float* inputs   = (const float*)d_in[0];   // [64, 1024] f32
  const int*   indexes  = (const int*)d_in[1];     // [64] i32
  const float* features = (const float*)d_in[2];   // [131072, 1024] f32

  float* sim      = (float*)d_out;                                  // [64, 131072]
  float* new_feat = (float*)d_out + (size_t)BATCH * N_BANK;         // [131072, 1024]
  float* Xn       = (float*)d_ws;                                   // 256 KB scratch

  // 1) normalize inputs
  normalize_rows<<<BATCH, 256, 0, stream>>>(inputs, Xn);

  // 2) copy the bank into the output (momentum update overwrites 64 rows later)
  const size_t n4 = (size_t)N_BANK * D_FEAT / 4;
  copy_bank<<<4096, 256, 0, stream>>>((const float4*)features, (float4*)new_feat, n4);

  // 3) similarity GEMM against the ORIGINAL bank (reads d_in, so order-safe)
  sim_gemm<<<N_BANK / 32, 256, 0, stream>>>(Xn, features, sim);

  // 4) sequential scatter update on the copied bank
  momentum_update<<<1, 1024, 0, stream>>>(Xn, indexes, new_feat);
}